// Retrace_82970178224358
// MI455X (gfx1250) — compile-verified
//
#include <hip/hip_runtime.h>
#include <cstdint>

// Retrace(λ=1) loss, B=8192 rows, T1=1025.
// One wave per row. Time parallelized inside the wave: two consecutive affine
// steps per lane, composed locally, then a wave32 Kogge-Stone suffix scan.
// Global->LDS streaming via gfx1250 async copies, 4-deep buffer pipeline
// (3 tiles in flight) sized from the HBM latency-bandwidth product.

#define GAMMA_F 0.99f
#define B_ROWS  8192
#define T1_LEN  1025
#define TILE    64          // time-steps per tile (2 per lane)
#define NTILES  16          // 16*64 = 1024 loss terms (t = 0..1023)
#define NBUF    4           // buffers; 3 tiles in flight
#define WPB     8           // waves per block

__device__ __forceinline__ void async_ld(const float* g, float* l) {
  // GLOBAL_LOAD_ASYNC_TO_LDS_B32: per-lane 4B copy, memory -> LDS (ASYNCcnt).
  uint32_t lds = (uint32_t)(uintptr_t)l;     // low 32 bits = LDS offset
  uint64_t ga  = (uint64_t)(uintptr_t)g;
  asm volatile("global_load_async_to_lds_b32 %0, %1, off"
               :: "v"(lds), "v"(ga) : "memory");
}

// Issue one full tile: 6 arrays x 64 floats, two coalesced 128B transactions
// per array (lane j -> slots j and j+32).
#define ISSUE_TILE(buf)                                                        \
  do {                                                                         \
    async_ld(pq,      &sm[w][buf][0][lane]);                                   \
    async_ld(pq + 32, &sm[w][buf][0][lane + 32]);                              \
    async_ld(pr,      &sm[w][buf][1][lane]);                                   \
    async_ld(pr + 32, &sm[w][buf][1][lane + 32]);                              \
    async_ld(pe,      &sm[w][buf][2][lane]);                                   \
    async_ld(pe + 32, &sm[w][buf][2][lane + 32]);                              \
    async_ld(pt,      &sm[w][buf][3][lane]);                                   \
    async_ld(pt + 32, &sm[w][buf][3][lane + 32]);                              \
    async_ld(pp,      &sm[w][buf][4][lane]);                                   \
    async_ld(pp + 32, &sm[w][buf][4][lane + 32]);                              \
    async_ld(pb,      &sm[w][buf][5][lane]);                                   \
    async_ld(pb + 32, &sm[w][buf][5][lane + 32]);                              \
  } while (0)

#define DEC_PTRS() do { pq -= TILE; pr -= TILE; pe -= TILE;                    \
                        pt -= TILE; pp -= TILE; pb -= TILE; } while (0)

__global__ __launch_bounds__(256) void retrace_scan_kernel(
    const float* __restrict__ Q,   const float* __restrict__ eQ,
    const float* __restrict__ tQ,  const float* __restrict__ R,
    const float* __restrict__ tpp, const float* __restrict__ bpp,
    float* __restrict__ partials)
{
  __shared__ float sm[WPB][NBUF][6][TILE];   // 48 KB, wave-private buffers
  __shared__ float wsum[WPB];

  const int lane = threadIdx.x & 31;
  const int w    = threadIdx.x >> 5;
  const int row  = blockIdx.x * WPB + w;     // 1024 blocks * 8 waves = 8192 rows
  const size_t rowoff = (size_t)row * T1_LEN;

  // Per-lane pointers for slot-j loads of tile k: element index = 64k + j + delta.
  const int top = (NTILES - 1) * TILE;       // 960
  const float* pq = Q   + rowoff + top + lane;       // delta 0 : Q[t]
  const float* pr = R   + rowoff + top + lane + 1;   // delta 1 : r[t+1]
  const float* pe = eQ  + rowoff + top + lane + 2;   // delta 2 : eQ[t+2]
  const float* pt = tQ  + rowoff + top + lane + 2;
  const float* pp = tpp + rowoff + top + lane + 2;
  const float* pb = bpp + rowoff + top + lane + 2;

  // ---- Prologue: issue tiles 15 (clamped tail), 14, 13 ----
  {
    const int buf = (NTILES - 1) & (NBUF - 1);
    async_ld(pq,      &sm[w][buf][0][lane]);
    async_ld(pq + 32, &sm[w][buf][0][lane + 32]);
    async_ld(pr,      &sm[w][buf][1][lane]);
    async_ld(pr + 32, &sm[w][buf][1][lane + 32]);
    async_ld(pe,      &sm[w][buf][2][lane]);
    async_ld(pt,      &sm[w][buf][3][lane]);
    async_ld(pp,      &sm[w][buf][4][lane]);
    async_ld(pb,      &sm[w][buf][5][lane]);
    // second half of the delta-2 arrays: index 994+lane, clamp 1025 -> 1024
    // (the clamped value is never used: that lane's 2nd step is identity).
    int idx2 = top + 32 + lane + 2;
    if (idx2 > 1024) idx2 = 1024;
    async_ld(eQ  + rowoff + idx2, &sm[w][buf][2][lane + 32]);
    async_ld(tQ  + rowoff + idx2, &sm[w][buf][3][lane + 32]);
    async_ld(tpp + rowoff + idx2, &sm[w][buf][4][lane + 32]);
    async_ld(bpp + rowoff + idx2, &sm[w][buf][5][lane + 32]);
  }
  DEC_PTRS(); ISSUE_TILE((NTILES - 2) & (NBUF - 1));   // tile 14
  DEC_PTRS(); ISSUE_TILE((NTILES - 3) & (NBUF - 1));   // tile 13

  float carry = Q[rowoff + 1024];            // G[1023] seed
  float acc = 0.0f;

  for (int k = NTILES - 1; k >= 0; --k) {
    const int buf = k & (NBUF - 1);
    if (k >= 3) {
      DEC_PTRS();
      ISSUE_TILE((k - 3) & (NBUF - 1));      // keep 3 tiles in flight
      asm volatile("s_wait_asynccnt 36" ::: "memory");   // tile k resident
    } else if (k == 2) {
      asm volatile("s_wait_asynccnt 24" ::: "memory");
    } else if (k == 1) {
      asm volatile("s_wait_asynccnt 12" ::: "memory");
    } else {
      asm volatile("s_wait_asynccnt 0"  ::: "memory");
    }

    // Lane j consumes time-steps t0 = 64k+2j and t1 = t0+1 (8B-aligned,
    // bank-conflict-free ds_load_b64 reads).
    const float2 qv = *(const float2*)&sm[w][buf][0][2 * lane];
    const float2 rv = *(const float2*)&sm[w][buf][1][2 * lane];
    const float2 ev = *(const float2*)&sm[w][buf][2][2 * lane];
    const float2 tv = *(const float2*)&sm[w][buf][3][2 * lane];
    const float2 pv = *(const float2*)&sm[w][buf][4][2 * lane];
    const float2 bv = *(const float2*)&sm[w][buf][5][2 * lane];

    // f_t(x) = b + a*x ;  a = g*c,  b = r[t+1] + g*eQ[t+2] - a*tQ[t+2]
    float a0 = GAMMA_F * __expf(fminf(pv.x - bv.x, 0.0f));
    float b0 = __fmaf_rn(-a0, tv.x, __fmaf_rn(GAMMA_F, ev.x, rv.x));
    float a1 = GAMMA_F * __expf(fminf(pv.y - bv.y, 0.0f));
    float b1 = __fmaf_rn(-a1, tv.y, __fmaf_rn(GAMMA_F, ev.y, rv.y));
    if (k == NTILES - 1 && lane == 31) { a1 = 1.0f; b1 = 0.0f; } // t=1023: identity

    // Local pair compose: F_j = f_{t0} o f_{t1}
    float A  = a0 * a1;
    float Bv = __fmaf_rn(a0, b1, b0);

    // Inclusive suffix scan over lanes: F_j <- f over pairs j..31.
    #pragma unroll
    for (int d = 1; d < 32; d <<= 1) {
      const float Ap = __shfl_down(A,  d, 32);
      const float Bp = __shfl_down(Bv, d, 32);
      if (lane + d < 32) { Bv = __fmaf_rn(A, Bp, Bv); A = A * Ap; }
    }
    // Exclusive suffix (from lane j+1), identity for lane 31.
    float Ax = __shfl_down(A,  1, 32);
    float Bx = __shfl_down(Bv, 1, 32);
    if (lane == 31) { Ax = 1.0f; Bx = 0.0f; }

    const float x  = __fmaf_rn(Ax, carry, Bx);   // G[t0+2]
    const float G1 = __fmaf_rn(a1, x,  b1);      // G[t0+1]
    const float G0 = __fmaf_rn(a0, G1, b0);      // G[t0]
    const float d0 = qv.x - G0;
    const float d1 = qv.y - G1;
    acc = __fmaf_rn(d0, d0, acc);
    acc = __fmaf_rn(d1, d1, acc);
    carry = __shfl(G0, 0, 32);                   // G[64k] feeds lower tile
  }

  // Deterministic reduction: wave shuffle tree -> per-block partial.
  #pragma unroll
  for (int d = 16; d >= 1; d >>= 1) acc += __shfl_down(acc, d, 32);
  if (lane == 0) wsum[w] = acc;
  __syncthreads();
  if (threadIdx.x == 0) {
    float s = 0.0f;
    #pragma unroll
    for (int i = 0; i < WPB; ++i) s += wsum[i];
    partials[blockIdx.x] = s;
  }
}

__global__ __launch_bounds__(256) void retrace_finalize_kernel(
    const float* __restrict__ partials, float* __restrict__ out)
{
  __shared__ float sh[256];
  float s = 0.0f;
  for (int i = threadIdx.x; i < 1024; i += 256) s += partials[i];
  sh[threadIdx.x] = s;
  __syncthreads();
  for (int stride = 128; stride > 0; stride >>= 1) {
    if ((int)threadIdx.x < stride) sh[threadIdx.x] += sh[threadIdx.x + stride];
    __syncthreads();
  }
  if (threadIdx.x == 0)
    out[0] = sh[0] * (1.0f / ((float)B_ROWS * 1024.0f));
}

extern "C" void kernel_launch(void* const* d_in, const int* in_sizes, int n_in,
                              void* d_out, int out_size, void* d_ws, size_t ws_size,
                              hipStream_t stream) {
  (void)in_sizes; (void)n_in; (void)out_size; (void)ws_size;
  // setup_inputs order: Q, expected_target_Q, target_Q, rewards,
  //                     target_policy_probs, behaviour_policy_probs
  const float* Q   = (const float*)d_in[0];
  const float* eQ  = (const float*)d_in[1];
  const float* tQ  = (const float*)d_in[2];
  const float* R   = (const float*)d_in[3];
  const float* tpp = (const float*)d_in[4];
  const float* bpp = (const float*)d_in[5];
  float* partials  = (float*)d_ws;           // 1024 floats of scratch

  retrace_scan_kernel<<<B_ROWS / WPB, 256, 0, stream>>>(
      Q, eQ, tQ, R, tpp, bpp, partials);
  retrace_finalize_kernel<<<1, 256, 0, stream>>>(partials, (float*)d_out);
}